// SoftmaxScalerLoss_59631325938458
// MI455X (gfx1250) — compile-verified
//
#include <hip/hip_runtime.h>
#include <cstdint>
#include <math.h>

// ---------------------------------------------------------------------------
// Masked-mean cross entropy, B*S independent rows of V fp32 logits.
// Memory-bound: 1.05 GB read once -> ~45us floor at 23.3 TB/s.
// CDNA5 paths: global_load_async_to_lds_b128 (4-deep ASYNCcnt pipeline,
// non-temporal), s_wait_asynccnt/s_wait_dscnt, wave32 shuffle reductions.
// ---------------------------------------------------------------------------

#define PADDING_IDX 0
#define ROW_THREADS 320                   // 10 wave32 per row-block
#define ROW_WAVES   (ROW_THREADS / 32)
#define TILE_FLOATS (ROW_THREADS * 4)     // 1280 floats: one b128 per lane
#define TILE_BYTES  (TILE_FLOATS * 4)     // 5120 B per tile; V=32000 -> 25 tiles
#define NBUF        4                     // async pipeline depth (20 KB LDS)

__device__ __forceinline__ void online_update(float& m, float& s, float x) {
  float nm = fmaxf(m, x);
  s = s * __expf(m - nm) + __expf(x - nm);
  m = nm;
}

__global__ __launch_bounds__(ROW_THREADS)
void ce_row_kernel(const float* __restrict__ logits,
                   const int*   __restrict__ targets,
                   float*       __restrict__ rownll,
                   int V, int nt) {
  __shared__ __align__(16) float buf[NBUF * TILE_FLOATS];   // 4-deep, 20 KB
  __shared__ float sm[ROW_WAVES], ss[ROW_WAVES];

  const int tid = threadIdx.x;
  const int row = blockIdx.x;
  const float* rowp = logits + (size_t)row * (size_t)V;

  // Escape the LDS buffer: written only by the async DMA engine, so the
  // optimizer must not reason about (or undef-fold) its contents.
  float* bp = buf;
  asm volatile("" : "+v"(bp));
  const unsigned lds_base = (unsigned)(uintptr_t)bp;        // low 32b = LDS offset
  const unsigned lane_b   = (unsigned)(tid * 16);

  float m = -INFINITY, s = 0.0f;

  auto issue_tile = [&](int tile) {
    unsigned lds  = lds_base + (unsigned)(tile & (NBUF - 1)) * TILE_BYTES + lane_b;
    unsigned voff = (unsigned)tile * TILE_BYTES + lane_b;
    // GVS mode: mem = SGPR64 base + VGPR32 offset; LDS dst offset in VGPR.
    // Non-temporal: 1 GB stream, zero reuse -> don't thrash the 192 MB L2.
    asm volatile("global_load_async_to_lds_b128 %0, %1, %2 offset:0 th:TH_LOAD_NT"
                 :: "v"(lds), "v"(voff), "s"(rowp) : "memory");
  };

  auto process_tile = [&](int tile) {
    const float4 v = *reinterpret_cast<const float4*>(
        bp + (tile & (NBUF - 1)) * TILE_FLOATS + tid * 4);
    float m4 = fmaxf(fmaxf(v.x, v.y), fmaxf(v.z, v.w));
    float nm = fmaxf(m, m4);
    s = s * __expf(m - nm)
        + __expf(v.x - nm) + __expf(v.y - nm)
        + __expf(v.z - nm) + __expf(v.w - nm);
    m = nm;
  };

  if (nt >= NBUF) {
    // Prologue: fill the 4-deep pipeline.
    issue_tile(0); issue_tile(1); issue_tile(2); issue_tile(3);

    // Steady state: branchless wait (drain to 3 outstanding), consume, refill.
    for (int k = 0; k + NBUF < nt; ++k) {
      asm volatile("s_wait_asynccnt 0x3" ::: "memory");
      process_tile(k);
      asm volatile("s_wait_dscnt 0x0" ::: "memory");  // our LDS reads retired
      issue_tile(k + NBUF);                           // overwrite oldest buffer
    }
    // Epilogue: drain remaining 4 tiles with descending wait targets.
    asm volatile("s_wait_asynccnt 0x3" ::: "memory"); process_tile(nt - 4);
    asm volatile("s_wait_asynccnt 0x2" ::: "memory"); process_tile(nt - 3);
    asm volatile("s_wait_asynccnt 0x1" ::: "memory"); process_tile(nt - 2);
    asm volatile("s_wait_asynccnt 0x0" ::: "memory"); process_tile(nt - 1);

    // Tail (V not a multiple of TILE_FLOATS) — not executed for V = 32000.
    for (int i = nt * TILE_FLOATS + tid; i < V; i += ROW_THREADS)
      online_update(m, s, rowp[i]);
  } else {
    // Degenerate small-V fallback: plain streaming loads.
    for (int i = tid; i < V; i += ROW_THREADS)
      online_update(m, s, rowp[i]);
  }

  // wave32 tree-merge of (max, sum) pairs
  #pragma unroll
  for (int off = 16; off > 0; off >>= 1) {
    float om = __shfl_xor(m, off);
    float os = __shfl_xor(s, off);
    float nm = fmaxf(m, om);
    s = s * __expf(m - nm) + os * __expf(om - nm);
    m = nm;
  }
  const int lane = tid & 31, wid = tid >> 5;
  if (lane == 0) { sm[wid] = m; ss[wid] = s; }
  __syncthreads();

  if (tid == 0) {
    float M = sm[0], S = ss[0];
    #pragma unroll
    for (int w = 1; w < ROW_WAVES; ++w) {
      float nm = fmaxf(M, sm[w]);
      S = S * __expf(M - nm) + ss[w] * __expf(sm[w] - nm);
      M = nm;
    }
    const int tgt = targets[row];
    const float xt = rowp[tgt];
    const float nll = M + __logf(S) - xt;               // logsumexp - x[target]
    rownll[row] = (tgt != PADDING_IDX) ? nll : 0.0f;
  }
}

// Deterministic fixed-order final reduction (no float atomics).
__global__ __launch_bounds__(256)
void ce_finalize_kernel(const float* __restrict__ rownll,
                        const int*   __restrict__ targets,
                        const float* __restrict__ scale,
                        float*       __restrict__ out,
                        int nrows) {
  __shared__ float s_sum[256];
  __shared__ float s_cnt[256];
  const int tid = threadIdx.x;
  float sum = 0.0f, cnt = 0.0f;
  for (int i = tid; i < nrows; i += 256) {
    sum += rownll[i];
    cnt += (targets[i] != PADDING_IDX) ? 1.0f : 0.0f;
  }
  s_sum[tid] = sum; s_cnt[tid] = cnt;
  __syncthreads();
  #pragma unroll
  for (int off = 128; off > 0; off >>= 1) {
    if (tid < off) { s_sum[tid] += s_sum[tid + off]; s_cnt[tid] += s_cnt[tid + off]; }
    __syncthreads();
  }
  if (tid == 0) out[0] = (s_sum[0] / s_cnt[0]) / scale[0];
}

extern "C" void kernel_launch(void* const* d_in, const int* in_sizes, int n_in,
                              void* d_out, int out_size, void* d_ws, size_t ws_size,
                              hipStream_t stream) {
  const float* logits  = (const float*)d_in[0];   // [B, S, V] fp32
  const int*   targets = (const int*)d_in[1];     // [B, S] int
  const float* scale   = (const float*)d_in[2];   // scalar
  float*       out     = (float*)d_out;           // scalar loss

  const int nrows = in_sizes[1];                  // B*S = 8192
  const int V     = in_sizes[0] / nrows;          // 32000
  const int nt    = V / TILE_FLOATS;              // 25 full tiles per row

  float* rownll = (float*)d_ws;                   // nrows floats of scratch

  ce_row_kernel<<<nrows, ROW_THREADS, 0, stream>>>(logits, targets, rownll, V, nt);
  ce_finalize_kernel<<<1, 256, 0, stream>>>(rownll, targets, scale, out, nrows);
}